// LatticeLossMix_85633057948137
// MI455X (gfx1250) — compile-verified
//
#include <hip/hip_runtime.h>

typedef __attribute__((ext_vector_type(2))) float v2f;
typedef __attribute__((ext_vector_type(8))) float v8f;

#define BATCH 32
#define TT 512
#define UU 512
#define CC 8
#define SS 8

__device__ __forceinline__ float fast_exp2(float x) {
#if defined(__HIP_DEVICE_COMPILE__) && __has_builtin(__builtin_amdgcn_exp2f)
    return __builtin_amdgcn_exp2f(x);      // v_exp_f32 (base-2)
#else
    return exp2f(x);
#endif
}
__device__ __forceinline__ float fast_log2(float x) {
#if defined(__HIP_DEVICE_COMPILE__) && __has_builtin(__builtin_amdgcn_logf)
    return __builtin_amdgcn_logf(x);       // v_log_f32 (base-2)
#else
    return log2f(x);
#endif
}

// Kernel 1: per-(batch, t-bin, u-bin) rectangle, accumulate per-class
// sum of -log softmax over all cells of the rectangle. Output: part[bid][c].
__global__ __launch_bounds__(256) void lattice_part_kernel(
    const float* __restrict__ logits,
    const int*   __restrict__ tlen,    // frame_label_length  [B,S] (t bands)
    const int*   __restrict__ ulen,    // frame_tlabel_length [B,S] (u bands)
    float*       __restrict__ part)    // [B*S*S, C]
{
    const int bid = blockIdx.x;
    const int b = bid >> 6;
    const int i = (bid >> 3) & 7;
    const int j = bid & 7;
    const int tid = threadIdx.x;
    const int tx = tid & 15;
    const int ty = tid >> 4;

    // bin boundaries from cumulative lengths (uniform scalar loads)
    int t0 = 0, t1 = 0, u0 = 0, u1 = 0;
    #pragma unroll
    for (int k = 0; k < SS; ++k) {
        int Lt = tlen[b * SS + k];
        int Lu = ulen[b * SS + k];
        if (k < i)  t0 += Lt;
        if (k <= i) t1 += Lt;
        if (k < j)  u0 += Lu;
        if (k <= j) u1 += Lu;
    }

    const float KLN2   = 0.69314718055994530942f;
    const float KLOG2E = 1.44269504088896340736f;
    const float CLIP   = 18.420680743952367f;   // -log(1e-8)

    float acc[CC];
    #pragma unroll
    for (int c = 0; c < CC; ++c) acc[c] = 0.0f;

    const size_t bbase = (size_t)b * TT * UU * CC;
    for (int t = t0 + ty; t < t1; t += 16) {
        const float* row = logits + bbase + (size_t)t * (UU * CC);
        for (int u = u0 + tx; u < u1; u += 16) {
            const float4* p = reinterpret_cast<const float4*>(row + u * CC);
            float4 lo = p[0];
            float4 hi = p[1];
            float v[CC] = {lo.x, lo.y, lo.z, lo.w, hi.x, hi.y, hi.z, hi.w};

            float m = v[0];
            #pragma unroll
            for (int c = 1; c < CC; ++c) m = fmaxf(m, v[c]);
            float ml = m * KLOG2E;
            float s = 0.0f;
            #pragma unroll
            for (int c = 0; c < CC; ++c)
                s += fast_exp2(__builtin_fmaf(v[c], KLOG2E, -ml));
            // base = logsumexp = m + ln2*log2(s); neglog_c = base - v[c]
            float base = __builtin_fmaf(KLN2, fast_log2(s), m);
            #pragma unroll
            for (int c = 0; c < CC; ++c)
                acc[c] += fminf(base - v[c], CLIP);
        }
    }

    // Stage the 256x8 accumulator matrix in LDS, reduce with WMMA.
    __shared__ float stage[CC][256];
    #pragma unroll
    for (int c = 0; c < CC; ++c) stage[c][tid] = acc[c];
    __syncthreads();

    if (tid < 32) {   // wave 0, EXEC all ones (uniform branch)
        const int lane  = tid;
        const int row   = lane & 15;          // A-matrix row m = lane % 16
        const int hi    = lane >> 4;          // lanes 16-31 carry K=2,3
        const int rowc  = row & 7;            // clamp address; mask by select
        const float rv  = (row < CC) ? 1.0f : 0.0f;

        v8f d = {0.f, 0.f, 0.f, 0.f, 0.f, 0.f, 0.f, 0.f};
        v2f ones; ones.x = 1.0f; ones.y = 1.0f;

        // D[m,n] = sum_k A[m,k]*1 + C[m,n]  -> row sums, B-layout independent.
        for (int r = 0; r < 64; ++r) {
            int k = 4 * r + 2 * hi;
            v2f a;
            a.x = stage[rowc][k]     * rv;
            a.y = stage[rowc][k + 1] * rv;
            d = __builtin_amdgcn_wmma_f32_16x16x4_f32(
                    false, a, false, ones, (short)0, d, false, false);
        }

        if (tid == 0) {
            float* o = part + (size_t)bid * CC;
            #pragma unroll
            for (int c = 0; c < CC; ++c) o[c] = d[c];  // D[M=c, N=0] for lane 0
        }
    }
}

// Kernel 2: prefix over bin grid, gather labels, loss = sum(val/count)/B.
__global__ __launch_bounds__(32) void lattice_final_kernel(
    const float* __restrict__ part,
    const int*   __restrict__ label,
    const int*   __restrict__ tlen,
    const int*   __restrict__ ulen,
    float*       __restrict__ out)
{
    __shared__ float red[BATCH];
    const int b = threadIdx.x;
    float loss = 0.0f;
    if (b < BATCH) {
        int tb[SS + 1], ub[SS + 1];
        tb[0] = 0; ub[0] = 0;
        #pragma unroll
        for (int k = 0; k < SS; ++k) {
            tb[k + 1] = tb[k] + tlen[b * SS + k];
            ub[k + 1] = ub[k] + ulen[b * SS + k];
        }
        float sq[CC];
        #pragma unroll
        for (int c = 0; c < CC; ++c) sq[c] = 0.0f;

        for (int s = 0; s < SS; ++s) {
            int cl = label[b * SS + s] & (CC - 1);
            float prev = sq[cl];
            // grow the prefix square by its L-shaped rim (row i=s, col j=s)
            for (int jj = 0; jj <= s; ++jj) {
                const float* p = part + (((size_t)(b * SS + s)) * SS + jj) * CC;
                #pragma unroll
                for (int c = 0; c < CC; ++c) sq[c] += p[c];
            }
            for (int ii = 0; ii < s; ++ii) {
                const float* p = part + (((size_t)(b * SS + ii)) * SS + s) * CC;
                #pragma unroll
                for (int c = 0; c < CC; ++c) sq[c] += p[c];
            }
            float val = sq[cl] - prev;   // R(te,ue) - R(ts,us) at class cl
            float cnt = (float)((long long)tb[s + 1] * ub[s + 1]
                              - (long long)tb[s]     * ub[s]);
            loss += val / cnt;
        }
    }
    red[threadIdx.x] = loss;
    __syncthreads();
    if (threadIdx.x == 0) {
        float tot = 0.0f;
        #pragma unroll
        for (int k = 0; k < BATCH; ++k) tot += red[k];
        out[0] = tot / (float)BATCH;
    }
}

extern "C" void kernel_launch(void* const* d_in, const int* in_sizes, int n_in,
                              void* d_out, int out_size, void* d_ws, size_t ws_size,
                              hipStream_t stream) {
    const float* logits = (const float*)d_in[0];
    const int*   label  = (const int*)d_in[1];
    const int*   tlen   = (const int*)d_in[2];   // frame_label_length  (t)
    const int*   ulen   = (const int*)d_in[3];   // frame_tlabel_length (u)
    float* part = (float*)d_ws;                  // B*S*S*C floats = 64 KB

    lattice_part_kernel<<<dim3(BATCH * SS * SS), 256, 0, stream>>>(
        logits, tlen, ulen, part);
    lattice_final_kernel<<<dim3(1), 32, 0, stream>>>(
        part, label, tlen, ulen, (float*)d_out);
}